// KGATLayer_52441550684531
// MI455X (gfx1250) — compile-verified
//
#include <hip/hip_runtime.h>

typedef __attribute__((ext_vector_type(2))) float v2f;
typedef __attribute__((ext_vector_type(8))) float v8f;

#define IN_DIM 128
#define OUT_DIM 128

// ---------------------------------------------------------------------------
// C[M x 128] = A[M x 128] * B[128 x 128], f32, via V_WMMA_F32_16X16X4_F32.
// One wave per 16-row strip; 8 accumulator tiles (16x128 of output) per wave;
// 32 K-steps of 4 -> 256 wmma ops per strip. Requires M % 16 == 0 (true for
// N=100000 and R=64); whole-wave guard keeps EXEC all-ones inside WMMA.
// A-matrix lane layout (ISA 7.12.2, 32-bit 16x4): lane m=l&15 is row M,
// lanes 0-15 hold K={k0,k0+1}, lanes 16-31 hold K={k0+2,k0+3}.
// C/D layout: VGPR v holds rows v (lanes 0-15) and v+8 (lanes 16-31).
// ---------------------------------------------------------------------------
__global__ void gemm128_wmma_f32(const float* __restrict__ A,
                                 const float* __restrict__ B,
                                 float* __restrict__ C, int M)
{
    const int lane  = threadIdx.x & 31;
    const int wave  = threadIdx.x >> 5;
    const int strip = blockIdx.x * (blockDim.x >> 5) + wave;
    const int row0  = strip * 16;
    if (row0 + 16 > M) return;              // whole-wave guard

    const int m  = lane & 15;               // row within strip (A) / col (B,C)
    const int kh = lane >> 4;               // K-half select

    v8f acc[8] = {};

    for (int k0 = 0; k0 < IN_DIM; k0 += 4) {
        const int ka = k0 + 2 * kh;
        const float* ap = A + (size_t)(row0 + m) * IN_DIM + ka;
        v2f av; av.x = ap[0]; av.y = ap[1];   // contiguous pair -> b64 load
        #pragma unroll
        for (int t = 0; t < 8; ++t) {
            v2f bv;
            bv.x = B[(size_t)ka * OUT_DIM       + t * 16 + m];
            bv.y = B[(size_t)(ka + 1) * OUT_DIM + t * 16 + m];
            acc[t] = __builtin_amdgcn_wmma_f32_16x16x4_f32(
                false, av, false, bv, (short)0, acc[t], false, false);
        }
    }

    #pragma unroll
    for (int t = 0; t < 8; ++t) {
        #pragma unroll
        for (int v = 0; v < 8; ++v) {
            C[(size_t)(row0 + v + 8 * kh) * OUT_DIM + t * 16 + m] = acc[t][v];
        }
    }
}

// ---------------------------------------------------------------------------
// o1[row] = dot(X[row], v1);  o2[row] = dot(X[row], v2).  One wave per row,
// float4 per lane (128 = 32 lanes * 4), wave32 butterfly reduce.
// ---------------------------------------------------------------------------
__global__ void row_dot2(const float* __restrict__ X,
                         const float* __restrict__ v1,
                         const float* __restrict__ v2,
                         float* __restrict__ o1,
                         float* __restrict__ o2, int M)
{
    const int lane = threadIdx.x & 31;
    const int row  = blockIdx.x * (blockDim.x >> 5) + (threadIdx.x >> 5);
    if (row >= M) return;
    const float4 xv = ((const float4*)(X + (size_t)row * OUT_DIM))[lane];
    const float4 w1 = ((const float4*)v1)[lane];
    const float4 w2 = ((const float4*)v2)[lane];
    float d1 = xv.x * w1.x + xv.y * w1.y + xv.z * w1.z + xv.w * w1.w;
    float d2 = xv.x * w2.x + xv.y * w2.y + xv.z * w2.z + xv.w * w2.w;
    #pragma unroll
    for (int off = 16; off; off >>= 1) {
        d1 += __shfl_xor(d1, off, 32);
        d2 += __shfl_xor(d2, off, 32);
    }
    if (lane == 0) { o1[row] = d1; o2[row] = d2; }
}

// ---------------------------------------------------------------------------
// Zero output / segment sums, reset encoded global max.
// ---------------------------------------------------------------------------
__global__ void init_kernel(float* __restrict__ out, float* __restrict__ att_sum,
                            unsigned* __restrict__ enc_max, size_t out_n, int n)
{
    size_t i = (size_t)blockIdx.x * blockDim.x + threadIdx.x;
    if (i < out_n) out[i] = 0.0f;
    if (i < (size_t)n) att_sum[i] = 0.0f;
    if (i == 0) *enc_max = 0u;
}

// Order-preserving float<->uint encoding for atomicMax over signed floats.
__device__ __forceinline__ unsigned enc_f32(float f) {
    unsigned u = __float_as_uint(f);
    return (u & 0x80000000u) ? ~u : (u | 0x80000000u);
}
__device__ __forceinline__ float dec_f32(unsigned u) {
    return __uint_as_float((u & 0x80000000u) ? (u & 0x7FFFFFFFu) : ~u);
}

// ---------------------------------------------------------------------------
// att[e] = leaky_relu(s1[src]+s2[type]+s3[dst]); wave-reduced global max.
// ---------------------------------------------------------------------------
__global__ void edge_logits(const int* __restrict__ src, const int* __restrict__ dst,
                            const int* __restrict__ et,
                            const float* __restrict__ s1, const float* __restrict__ s2,
                            const float* __restrict__ s3,
                            float* __restrict__ att, unsigned* __restrict__ enc_max,
                            int E)
{
    const int e = blockIdx.x * blockDim.x + threadIdx.x;
    float v = -3.4e38f;
    if (e < E) {
        float t = s1[src[e]] + s2[et[e]] + s3[dst[e]];
        v = (t > 0.0f) ? t : 0.2f * t;      // leaky_relu, slope 0.2
        att[e] = v;
    }
    #pragma unroll
    for (int off = 16; off; off >>= 1) {
        float o = __shfl_xor(v, off, 32);
        v = (v > o) ? v : o;
    }
    if ((threadIdx.x & 31) == 0) atomicMax(enc_max, enc_f32(v));
}

// ---------------------------------------------------------------------------
// att[e] = exp(att[e]-max); att_sum[dst] += att[e]  (segment sum via atomics)
// ---------------------------------------------------------------------------
__global__ void edge_exp(const int* __restrict__ dst,
                         float* __restrict__ att, float* __restrict__ att_sum,
                         const unsigned* __restrict__ enc_max, int E)
{
    const int e = blockIdx.x * blockDim.x + threadIdx.x;
    if (e >= E) return;
    const float mx = dec_f32(*enc_max);
    const float v = expf(att[e] - mx);
    att[e] = v;
    atomicAdd(att_sum + dst[e], v);
}

// ---------------------------------------------------------------------------
// out[dst] += (att[e]/(att_sum[dst]+1e-10)) * xt[src].  One wave per edge,
// float4 per lane gather + 4 global f32 atomic adds. This is the bandwidth-
// bound hot loop (~1 GB total -> ~44us at 23.3 TB/s).
// ---------------------------------------------------------------------------
__global__ void edge_scatter(const int* __restrict__ src, const int* __restrict__ dst,
                             const float* __restrict__ att,
                             const float* __restrict__ att_sum,
                             const float* __restrict__ xt,
                             float* __restrict__ out, int E)
{
    const int lane = threadIdx.x & 31;
    const int e = blockIdx.x * (blockDim.x >> 5) + (threadIdx.x >> 5);
    if (e >= E) return;
    const int s = src[e], d = dst[e];
    const float coef = att[e] / (att_sum[d] + 1e-10f);
    const float4 xs = ((const float4*)(xt + (size_t)s * OUT_DIM))[lane];
    float* op = out + (size_t)d * OUT_DIM + lane * 4;
    atomicAdd(op + 0, coef * xs.x);
    atomicAdd(op + 1, coef * xs.y);
    atomicAdd(op + 2, coef * xs.z);
    atomicAdd(op + 3, coef * xs.w);
}

// ---------------------------------------------------------------------------
extern "C" void kernel_launch(void* const* d_in, const int* in_sizes, int n_in,
                              void* d_out, int out_size, void* d_ws, size_t ws_size,
                              hipStream_t stream)
{
    const float* x       = (const float*)d_in[0];
    const int*   ei      = (const int*)d_in[1];   // [2, E]
    const int*   et      = (const int*)d_in[2];   // [E]
    const float* W       = (const float*)d_in[3];
    const float* W_r     = (const float*)d_in[4];
    const float* a       = (const float*)d_in[5]; // [3*128]
    const float* rel_emb = (const float*)d_in[6]; // [R, 128]

    const int E = in_sizes[2];
    const int N = in_sizes[0] / IN_DIM;
    const int R = in_sizes[6] / IN_DIM;

    const int* src = ei;
    const int* dst = ei + E;

    // Workspace layout (floats)
    float* ws = (float*)d_ws;
    float* xt      = ws;  ws += (size_t)N * OUT_DIM;   // projected nodes
    float* rel_t   = ws;  ws += (size_t)R * OUT_DIM;   // projected relations
    float* s1      = ws;  ws += N;
    float* s3      = ws;  ws += N;
    float* s2      = ws;  ws += R;
    float* s2d     = ws;  ws += R;                     // scratch (dup dot)
    float* att     = ws;  ws += E;
    float* att_sum = ws;  ws += N;
    unsigned* enc_max = (unsigned*)ws;

    float* out = (float*)d_out;

    // 1) node projection + relation projection (WMMA f32 GEMMs)
    gemm128_wmma_f32<<<(N + 127) / 128, 256, 0, stream>>>(x, W, xt, N);
    gemm128_wmma_f32<<<(R + 127) / 128, 256, 0, stream>>>(rel_emb, W_r, rel_t, R);

    // 2) factorized attention scalars: s1 = xt*a1, s3 = xt*a3, s2 = rel_t*a2
    row_dot2<<<(N + 7) / 8, 256, 0, stream>>>(xt, a, a + 2 * OUT_DIM, s1, s3, N);
    row_dot2<<<(R + 7) / 8, 256, 0, stream>>>(rel_t, a + OUT_DIM, a + OUT_DIM, s2, s2d, R);

    // 3) init accumulators
    const size_t out_n = (size_t)N * OUT_DIM;
    init_kernel<<<(unsigned)((out_n + 255) / 256), 256, 0, stream>>>(out, att_sum, enc_max, out_n, N);

    // 4) logits + global max
    edge_logits<<<(E + 255) / 256, 256, 0, stream>>>(src, dst, et, s1, s2, s3, att, enc_max, E);

    // 5) exp + per-dst sums
    edge_exp<<<(E + 255) / 256, 256, 0, stream>>>(dst, att, att_sum, enc_max, E);

    // 6) weighted scatter of messages
    edge_scatter<<<(E + 7) / 8, 256, 0, stream>>>(src, dst, att, att_sum, xt, out, E);
}